// HyperConv2d_54546084659204
// MI455X (gfx1250) — compile-verified
//
#include <hip/hip_runtime.h>
#include <cstdint>

typedef __attribute__((ext_vector_type(2))) float v2f;
typedef __attribute__((ext_vector_type(8))) float v8f;

#define B_      32
#define C_IN    64
#define C_OUT   64
#define H_      128
#define W_      128
#define HYP_    64
#define KTOT    (C_IN * 9)        // 576 = reduction length (c*9 + kh*3 + kw)
#define KUNITS  (C_OUT * KTOT)    // 36864 hypernet-generated weights per sample

#define TILE_W  64                // pixels per workgroup (half an image row)
#define COLS    (TILE_W + 2)      // 66: halo columns
#define ROWS    3                 // 3 input rows feed one output row
#define XS_N    (C_IN * ROWS * COLS) // 12672 floats = 50688 B of LDS

#if defined(__AMDGCN__) && __has_builtin(__builtin_amdgcn_global_load_async_to_lds_b32)
#define USE_ASYNC_G2L 1
typedef __attribute__((address_space(1))) int gas_int;   // global
typedef __attribute__((address_space(3))) int las_int;   // LDS
#else
#define USE_ASYNC_G2L 0
#endif

// ---------------------------------------------------------------------------
// Hypernet: w[b][o][c][kh][kw] = tanh(hyp[b,:] . Wk[u,:] + bk[u]),
// u = o*576 + c*9 + kh*3 + kw  (matches reference reshape order).
// ---------------------------------------------------------------------------
__global__ void hyper_weights_kernel(const float* __restrict__ hyp,
                                     const float* __restrict__ Wk,
                                     const float* __restrict__ bk,
                                     float* __restrict__ wsW) {
    __shared__ float hs[HYP_];
    const int b = blockIdx.y;
    if (threadIdx.x < HYP_) hs[threadIdx.x] = hyp[b * HYP_ + threadIdx.x];
    __syncthreads();
    const int u = blockIdx.x * blockDim.x + threadIdx.x;   // grid sized exactly
    const float4* wr = (const float4*)(Wk + (size_t)u * HYP_);
    float s = bk[u];
#pragma unroll
    for (int i = 0; i < HYP_ / 4; ++i) {
        float4 w4 = wr[i];
        s = fmaf(hs[4 * i + 0], w4.x, s);
        s = fmaf(hs[4 * i + 1], w4.y, s);
        s = fmaf(hs[4 * i + 2], w4.z, s);
        s = fmaf(hs[4 * i + 3], w4.w, s);
    }
    wsW[(size_t)b * KUNITS + u] = tanhf(s);
}

__global__ void hyper_bias_kernel(const float* __restrict__ hyp,
                                  const float* __restrict__ Wb,
                                  const float* __restrict__ bb,
                                  float* __restrict__ wsB) {
    const int b = blockIdx.x, o = threadIdx.x;             // 32 x 64
    const float* h  = hyp + b * HYP_;
    const float* wr = Wb + o * HYP_;
    float s = bb[o];
#pragma unroll
    for (int i = 0; i < HYP_; ++i) s = fmaf(h[i], wr[i], s);
    wsB[b * C_OUT + o] = tanhf(s);
}

// ---------------------------------------------------------------------------
// Implicit-GEMM conv:  out[b] = W_b[64 x 576] * im2col(x_b)[576 x 16384]
// One workgroup = one sample, one output row h, 64 pixels (w0..w0+63).
// 4 waves; wave w -> (mt = w&1, ntq = w>>1): 32x32 output tile per wave
// (2 A fragments x 2 B fragments -> 4 WMMA per K-step of 4).
// LDS im2col bases advance via the recurrence base(k+4) = base(k) + (67|130),
// so no div/mod in the hot loop.
// ---------------------------------------------------------------------------
__global__ void __launch_bounds__(128)
hyperconv_wmma_kernel(const float* __restrict__ x,
                      const float* __restrict__ wsW,
                      const float* __restrict__ wsB,
                      float* __restrict__ out) {
    __shared__ float xs[XS_N];
    const int b    = blockIdx.y;
    const int tile = blockIdx.x;          // 0..255: h = tile/2, half-row = tile&1
    const int h    = tile >> 1;
    const int w0   = (tile & 1) * TILE_W;

    // Stage x slab with zero halo: 12672/128 = 99 elems/thread, zero divergence
    const size_t xb = (size_t)b * C_IN * H_ * W_;
    for (int i = threadIdx.x; i < XS_N; i += 128) {
        int c   = i / (ROWS * COLS);
        int rem = i - c * (ROWS * COLS);
        int r   = rem / COLS;             // 0..2  -> input row h-1+r
        int col = rem - r * COLS;         // 0..65 -> input col w0-1+col
        int hh = h - 1 + r;
        int ww = w0 - 1 + col;
        bool inb = ((unsigned)hh < (unsigned)H_) && ((unsigned)ww < (unsigned)W_);
#if USE_ASYNC_G2L
        if (inb) {
            __builtin_amdgcn_global_load_async_to_lds_b32(
                (gas_int*)(x + xb + ((size_t)c << 14) + (hh << 7) + ww),
                (las_int*)(xs + i), 0, 0);
        } else {
            xs[i] = 0.f;
        }
#else
        xs[i] = inb ? x[xb + ((size_t)c << 14) + (hh << 7) + ww] : 0.f;
#endif
    }
#if USE_ASYNC_G2L
    asm volatile("s_wait_asynccnt 0" ::: "memory");
#endif
    __syncthreads();

    const int wave  = threadIdx.x >> 5;
    const int lane  = threadIdx.x & 31;
    const int lhalf = lane >> 4;          // lane group: 0 -> K+{0,1}, 1 -> K+{2,3}
    const int l16   = lane & 15;
    const int mt    = wave & 1;           // M half: output rows [32*mt, 32*mt+32)
    const int ntq   = wave >> 1;          // N half: pixels [32*ntq, 32*ntq+32)

    // A fragments: rows m0 and m0+16, two consecutive K values each (b64 loads)
    const int m0 = 32 * mt + l16;
    const float* wrow0 = wsW + ((size_t)b * C_OUT + m0) * KTOT + 2 * lhalf;
    const float* wrow1 = wrow0 + 16 * KTOT;

    // B fragment LDS bases for k = 2*lhalf (base0) and k = 2*lhalf+1 (base1)
    const int ncol = 32 * ntq + l16;
    int base0 = ncol + 2 * lhalf;             // k=0 -> +0 ; k=2 -> kh0,kw2 -> +2
    int base1 = base0 + (lhalf ? 64 : 1);     // k=1 -> +1 ; k=3 -> kh1,kw0 -> +66
    int rmod  = lhalf ? 2 : 0;                // (k of base0) % 3

    v8f acc00 = {}, acc01 = {}, acc10 = {}, acc11 = {};

#pragma unroll 3
    for (int k0 = 0; k0 < KTOT; k0 += 4) {
        v2f a0 = *(const v2f*)(wrow0 + k0);
        v2f a1 = *(const v2f*)(wrow1 + k0);

        v2f b0; b0[0] = xs[base0];      b0[1] = xs[base1];
        v2f b1; b1[0] = xs[base0 + 16]; b1[1] = xs[base1 + 16];

        acc00 = __builtin_amdgcn_wmma_f32_16x16x4_f32(false, a0, false, b0, (short)0, acc00, false, false);
        acc01 = __builtin_amdgcn_wmma_f32_16x16x4_f32(false, a0, false, b1, (short)0, acc01, false, false);
        acc10 = __builtin_amdgcn_wmma_f32_16x16x4_f32(false, a1, false, b0, (short)0, acc10, false, false);
        acc11 = __builtin_amdgcn_wmma_f32_16x16x4_f32(false, a1, false, b1, (short)0, acc11, false, false);

        // base(k+4) = base(k) + 66 + d(k),  d(k) = (k%3==2 ? 64 : 1)
        int rnext = (rmod == 2) ? 0 : rmod + 1;      // (k+1) % 3
        base0 += (rmod  == 2) ? 130 : 67;
        base1 += (rnext == 2) ? 130 : 67;
        rmod = rnext;
    }

    // Epilogue: C/D layout (ISA 7.12.2): VGPR j -> row 8*lhalf + j, col = l16.
    const size_t ob = ((size_t)b * C_OUT) << 14;
    const int colbase = w0 + 32 * ntq + l16;
#pragma unroll
    for (int j = 0; j < 8; ++j) {
        const int row0 = 32 * mt + 8 * lhalf + j;
        const int row1 = row0 + 16;
        const float bia0 = wsB[b * C_OUT + row0];
        const float bia1 = wsB[b * C_OUT + row1];
        const size_t p0 = ob + ((size_t)row0 << 14) + (h << 7) + colbase;
        const size_t p1 = ob + ((size_t)row1 << 14) + (h << 7) + colbase;
        out[p0]      = acc00[j] + bia0;
        out[p0 + 16] = acc01[j] + bia0;
        out[p1]      = acc10[j] + bia1;
        out[p1 + 16] = acc11[j] + bia1;
    }
}

// ---------------------------------------------------------------------------
extern "C" void kernel_launch(void* const* d_in, const int* in_sizes, int n_in,
                              void* d_out, int out_size, void* d_ws, size_t ws_size,
                              hipStream_t stream) {
    const float* x   = (const float*)d_in[0];  // [32,64,128,128]
    const float* hyp = (const float*)d_in[1];  // [32,64]
    const float* Wk  = (const float*)d_in[2];  // [36864,64]
    const float* bk  = (const float*)d_in[3];  // [36864]
    const float* Wb  = (const float*)d_in[4];  // [64,64]
    const float* bb  = (const float*)d_in[5];  // [64]
    float* out = (float*)d_out;                // [32,64,128,128]

    float* wsW = (float*)d_ws;                         // 32*36864 floats
    float* wsB = wsW + (size_t)B_ * KUNITS;            // + 32*64 floats

    // 1) hypernet weights: grid 144*256 = 36864 exactly
    hyper_weights_kernel<<<dim3(KUNITS / 256, B_), 256, 0, stream>>>(hyp, Wk, bk, wsW);
    // 2) hypernet bias
    hyper_bias_kernel<<<B_, C_OUT, 0, stream>>>(hyp, Wb, bb, wsB);
    // 3) WMMA conv: 256 pixel-tiles per sample, 32 samples, 128 threads each
    hyperconv_wmma_kernel<<<dim3(256, B_), 128, 0, stream>>>(x, wsW, wsB, out);
}